// Helm_with_a_13572096655538
// MI455X (gfx1250) — compile-verified
//
#include <hip/hip_runtime.h>

// sin(pi * x) via hardware V_SIN_F32 (input unit = revolutions: sin(2*pi*s)),
// so sin(pi*x) = v_sin_f32(0.5 * x).  x in [0,1) -> argument in [0,0.5), well
// inside the hardware's exact-range.
__device__ __forceinline__ float sin_pi(float x) {
    return __builtin_amdgcn_sinf(0.5f * x);
}

typedef float v4f __attribute__((ext_vector_type(4)));

#define PI_F 3.14159265358979323846f

// Vector kernel: 4 rows per thread.
//   input  rows 4t..4t+3  -> two contiguous float4 loads (global_load_b128)
//   output rows 4t..4t+3  -> one float4 non-temporal store (global_store_b128 NT)
__global__ __launch_bounds__(256) void helm_vec4_kernel(
    const v4f* __restrict__ in4,   // [2*n4] float4 = interleaved (x0,x1) pairs
    const float* __restrict__ a,   // [1]
    v4f* __restrict__ out4,        // [n4]
    int n4)
{
    // Uniform scalar path: s_load_b32 + SGPR math for the coefficient.
    const float av   = a[0];
    const float coef = av * av - 2.0f * PI_F * PI_F;   // -pi^2 - (P*pi)^2 + a^2, P=1

    const int t = blockIdx.x * blockDim.x + threadIdx.x;
    if (t < n4) {
        // RT loads: let the 192MB L2 keep the input resident across replays.
        const v4f v0 = in4[2 * t + 0];   // rows 4t, 4t+1 : (x0,x1,x0,x1)
        const v4f v1 = in4[2 * t + 1];   // rows 4t+2, 4t+3

        v4f r;
        r.x = coef * sin_pi(v0.x) * sin_pi(v0.y);
        r.y = coef * sin_pi(v0.z) * sin_pi(v0.w);
        r.z = coef * sin_pi(v1.x) * sin_pi(v1.y);
        r.w = coef * sin_pi(v1.z) * sin_pi(v1.w);

        // Non-temporal store: output is write-once (validated later from HBM/L2
        // anyway); don't evict the reusable 128MB input from L2.
        __builtin_nontemporal_store(r, &out4[t]);
    }
}

// Scalar tail for n % 4 != 0 (not hit for N = 2^24, kept for generality).
__global__ void helm_tail_kernel(
    const float* __restrict__ in,  // [2*n] interleaved
    const float* __restrict__ a,
    float* __restrict__ out,       // [n]
    int start, int n)
{
    const float av   = a[0];
    const float coef = av * av - 2.0f * PI_F * PI_F;
    int i = start + blockIdx.x * blockDim.x + threadIdx.x;
    if (i < n) {
        const float x0 = in[2 * i + 0];
        const float x1 = in[2 * i + 1];
        out[i] = coef * sin_pi(x0) * sin_pi(x1);
    }
}

extern "C" void kernel_launch(void* const* d_in, const int* in_sizes, int n_in,
                              void* d_out, int out_size, void* d_ws, size_t ws_size,
                              hipStream_t stream)
{
    const float* in = (const float*)d_in[0];   // [N, 2] float32
    const float* a  = (const float*)d_in[1];   // [1]   float32
    float* out      = (float*)d_out;           // [N]   float32 (shape [N,1])

    const int n   = out_size;      // number of rows (16,777,216)
    const int n4  = n >> 2;        // float4 groups of 4 rows
    const int rem = n - (n4 << 2);

    if (n4 > 0) {
        const int threads = 256;
        const int blocks  = (n4 + threads - 1) / threads;
        helm_vec4_kernel<<<blocks, threads, 0, stream>>>(
            (const v4f*)in, a, (v4f*)out, n4);
    }
    if (rem > 0) {
        helm_tail_kernel<<<1, 64, 0, stream>>>(in, a, out, n4 << 2, n);
    }
}